// SkillRegistry_8581344657493
// MI455X (gfx1250) — compile-verified
//
#include <hip/hip_runtime.h>
#include <hip/hip_bf16.h>

#define BB 4
#define TT 1024
#define DD 2048
#define SKD 128
#define NSK 4096
#define TOPK 8

typedef __attribute__((ext_vector_type(2))) float v2f;
typedef __attribute__((ext_vector_type(8))) float v8f;

// ---------- K1: query[b] = h[b, T-1, :] @ ret_wq + ret_bq  (4 x 2048 x 128) ----------
__global__ void k_query(const float* __restrict__ h, const float* __restrict__ wq,
                        const float* __restrict__ bq, float* __restrict__ query) {
    __shared__ float hrow[DD];
    const int b = blockIdx.x;
    const float* hp = h + ((size_t)b * TT + (TT - 1)) * DD;
    for (int k = threadIdx.x; k < DD; k += blockDim.x) hrow[k] = hp[k];
    __syncthreads();
    const int n = threadIdx.x;           // blockDim == 128
    float acc = bq[n];
    for (int k = 0; k < DD; ++k) acc = fmaf(hrow[k], wq[k * SKD + n], acc);
    query[b * SKD + n] = acc;
}

// ---------- K2: keys = embeddings @ ret_wk + ret_bk  via V_WMMA_F32_16X16X4_F32 ----------
// 4096x128x128 fp32 GEMM. One wave per 16x16 output tile, K-loop of 32 WMMA steps.
__global__ void k_keys_wmma(const float* __restrict__ emb, const float* __restrict__ wk,
                            const float* __restrict__ bk, float* __restrict__ keys) {
    const int wave  = blockIdx.x * (blockDim.x >> 5) + (threadIdx.x >> 5);
    const int lane  = threadIdx.x & 31;
    const int half  = lane >> 4;         // 0: K pair {0,1}, 1: K pair {2,3}
    const int r     = lane & 15;
    const int mtile = wave >> 3;         // 256 row tiles
    const int ntile = wave & 7;          // 8 col tiles
    const int m = mtile * 16 + r;
    const int n = ntile * 16 + r;

    v8f c = {};
    const float* arow = emb + (size_t)m * SKD + 2 * half;            // A[r][k+2h], A[r][k+2h+1]
    const float* bcol = wk + (size_t)(2 * half) * SKD + n;           // B[k+2h][n], B[k+2h+1][n]
    #pragma unroll 4
    for (int k = 0; k < SKD; k += 4) {
        v2f a, bf;
        a.x  = arow[k];
        a.y  = arow[k + 1];
        bf.x = bcol[(size_t)k * SKD];
        bf.y = bcol[(size_t)(k + 1) * SKD];
        c = __builtin_amdgcn_wmma_f32_16x16x4_f32(false, a, false, bf,
                                                  (short)0, c, false, false);
    }
    const float bias = bk[n];
    #pragma unroll
    for (int i = 0; i < 8; ++i) {        // C/D: VGPR i -> M = i + 8*half, N = r
        const int row = mtile * 16 + half * 8 + i;
        keys[(size_t)row * SKD + n] = c[i] + bias;
    }
}

// ---------- K3: scores[b][s] = (query[b] . keys[s]) / sqrt(128) ----------
__global__ void k_scores(const float* __restrict__ keys, const float* __restrict__ query,
                         float* __restrict__ scores) {
    __shared__ float q[BB * SKD];
    for (int i = threadIdx.x; i < BB * SKD; i += blockDim.x) q[i] = query[i];
    __syncthreads();
    const int s = blockIdx.x * blockDim.x + threadIdx.x;   // 4096 total
    const float* krow = keys + (size_t)s * SKD;
    float a0 = 0.f, a1 = 0.f, a2 = 0.f, a3 = 0.f;
    for (int j = 0; j < SKD; ++j) {
        const float kv = krow[j];
        a0 = fmaf(kv, q[0 * SKD + j], a0);
        a1 = fmaf(kv, q[1 * SKD + j], a1);
        a2 = fmaf(kv, q[2 * SKD + j], a2);
        a3 = fmaf(kv, q[3 * SKD + j], a3);
    }
    const float inv = 0.0883883476483184406f;              // 1/sqrt(128)
    scores[0 * NSK + s] = a0 * inv;
    scores[1 * NSK + s] = a1 * inv;
    scores[2 * NSK + s] = a2 * inv;
    scores[3 * NSK + s] = a3 * inv;
}

// ---------- K4: top-8 (desc, lowest-index tie-break) + softmax + skill combine ----------
__global__ void k_topk_skill(const float* __restrict__ scores, const float* __restrict__ emb,
                             float* __restrict__ skill_out, float* __restrict__ skill_ws) {
    __shared__ float sc[NSK];
    __shared__ float rmax[256];
    __shared__ int   ridx[256];
    __shared__ int   tidx[TOPK];
    __shared__ float tval[TOPK];
    __shared__ float w[TOPK];
    const int b = blockIdx.x, tid = threadIdx.x;
    for (int s = tid; s < NSK; s += 256) sc[s] = scores[b * NSK + s];
    __syncthreads();
    for (int k = 0; k < TOPK; ++k) {
        float best = -__builtin_inff(); int bi = 0x7fffffff;
        for (int s = tid; s < NSK; s += 256) {
            const float v = sc[s];
            if (v > best || (v == best && s < bi)) { best = v; bi = s; }
        }
        rmax[tid] = best; ridx[tid] = bi;
        __syncthreads();
        for (int off = 128; off > 0; off >>= 1) {
            if (tid < off) {
                const float ov = rmax[tid + off]; const int oi = ridx[tid + off];
                if (ov > rmax[tid] || (ov == rmax[tid] && oi < ridx[tid])) {
                    rmax[tid] = ov; ridx[tid] = oi;
                }
            }
            __syncthreads();
        }
        if (tid == 0) { tidx[k] = ridx[0]; tval[k] = rmax[0]; sc[ridx[0]] = -__builtin_inff(); }
        __syncthreads();
    }
    if (tid == 0) {
        const float m = tval[0];                           // first pick is the max
        float e[TOPK]; float sum = 0.f;
        for (int k = 0; k < TOPK; ++k) { e[k] = __expf(tval[k] - m); sum += e[k]; }
        for (int k = 0; k < TOPK; ++k) w[k] = e[k] / sum;
    }
    __syncthreads();
    if (tid < SKD) {
        float acc = 0.f;
        for (int k = 0; k < TOPK; ++k)
            acc = fmaf(w[k], emb[(size_t)tidx[k] * SKD + tid], acc);
        skill_out[b * SKD + tid] = acc;                    // output #2
        skill_ws [b * SKD + tid] = acc;
    }
}

// ---------- K5: v[b][j] = skill[b] @ ad_wkv[:, D+j] + ad_bkv[D+j] ----------
__global__ void k_v(const float* __restrict__ skill, const float* __restrict__ wkv,
                    const float* __restrict__ bkv, float* __restrict__ v) {
    const int b = blockIdx.x >> 3;
    const int j = (blockIdx.x & 7) * 256 + threadIdx.x;    // 0..2047
    __shared__ float sk[SKD];
    if (threadIdx.x < SKD) sk[threadIdx.x] = skill[b * SKD + threadIdx.x];
    __syncthreads();
    float acc = bkv[DD + j];
    for (int i = 0; i < SKD; ++i)
        acc = fmaf(sk[i], wkv[(size_t)i * (2 * DD) + DD + j], acc);
    v[b * DD + j] = acc;
}

// ---------- K6: fout[b][n] = v[b] @ ad_wo[:, n]  (one pass over the 16MB ad_wo) ----------
__global__ void k_fout(const float* __restrict__ v, const float* __restrict__ wo,
                       float* __restrict__ fout) {
    const int n = blockIdx.x * 256 + threadIdx.x;          // 0..2047
    __shared__ float sv[BB * DD];                          // 32 KB
    for (int i = threadIdx.x; i < BB * DD; i += 256) sv[i] = v[i];
    __syncthreads();
    float a0 = 0.f, a1 = 0.f, a2 = 0.f, a3 = 0.f;
    for (int d = 0; d < DD; ++d) {
        if ((d & 63) == 0 && d + 64 < DD)
            __builtin_prefetch(&wo[(size_t)(d + 64) * DD + n], 0, 1);  // global_prefetch_b8
        const float wv = wo[(size_t)d * DD + n];
        a0 = fmaf(sv[0 * DD + d], wv, a0);
        a1 = fmaf(sv[1 * DD + d], wv, a1);
        a2 = fmaf(sv[2 * DD + d], wv, a2);
        a3 = fmaf(sv[3 * DD + d], wv, a3);
    }
    fout[0 * DD + n] = a0; fout[1 * DD + n] = a1;
    fout[2 * DD + n] = a2; fout[3 * DD + n] = a3;
}

// ---------- K7: h_executed[b,t,:] = fout[b,:]  (32 MB broadcast store) ----------
__global__ void k_broadcast(const float* __restrict__ fout, float4* __restrict__ out) {
    const size_t total = (size_t)BB * TT * DD / 4;
    size_t idx = (size_t)blockIdx.x * blockDim.x + threadIdx.x;
    const size_t stride = (size_t)gridDim.x * blockDim.x;
    const float4* f4 = (const float4*)fout;
    for (; idx < total; idx += stride) {
        const size_t f = idx * 4;
        const int b  = (int)(f / ((size_t)TT * DD));
        const int n4 = (int)((f % DD) >> 2);
        out[idx] = f4[b * (DD / 4) + n4];
    }
}

extern "C" void kernel_launch(void* const* d_in, const int* in_sizes, int n_in,
                              void* d_out, int out_size, void* d_ws, size_t ws_size,
                              hipStream_t stream) {
    const float* h      = (const float*)d_in[0];
    const float* emb    = (const float*)d_in[1];
    const float* ret_wq = (const float*)d_in[2];
    const float* ret_bq = (const float*)d_in[3];
    const float* ret_wk = (const float*)d_in[4];
    const float* ret_bk = (const float*)d_in[5];
    // d_in[6] ctrl_w, d_in[7] ctrl_b, d_in[8] gate, d_in[11] ad_wq are mathematically
    // dead: broadcast K/V => uniform softmax => attention output == V independent of q/hc.
    const float* ad_wkv = (const float*)d_in[9];
    const float* ad_bkv = (const float*)d_in[10];
    const float* ad_wo  = (const float*)d_in[12];

    float* out_exec  = (float*)d_out;                      // [B,T,D]
    float* out_skill = out_exec + (size_t)BB * TT * DD;    // [B,SKILL_DIM]

    float* ws       = (float*)d_ws;
    float* w_query  = ws; ws += BB * SKD;                  // 512
    float* w_keys   = ws; ws += (size_t)NSK * SKD;         // 524288
    float* w_scores = ws; ws += BB * NSK;                  // 16384
    float* w_skill  = ws; ws += BB * SKD;                  // 512
    float* w_v      = ws; ws += BB * DD;                   // 8192
    float* w_fout   = ws;                                  // 8192   (~2.2 MB total)

    k_query     <<<BB, SKD, 0, stream>>>(h, ret_wq, ret_bq, w_query);
    k_keys_wmma <<<(NSK / 16) * (SKD / 16) / 8, 256, 0, stream>>>(emb, ret_wk, ret_bk, w_keys);
    k_scores    <<<NSK / 256, 256, 0, stream>>>(w_keys, w_query, w_scores);
    k_topk_skill<<<BB, 256, 0, stream>>>(w_scores, emb, out_skill, w_skill);
    k_v         <<<BB * 8, 256, 0, stream>>>(w_skill, ad_wkv, ad_bkv, w_v);
    k_fout      <<<DD / 256, 256, 0, stream>>>(w_v, ad_wo, w_fout);
    k_broadcast <<<4096, 256, 0, stream>>>(w_fout, (float4*)out_exec);
}